// TransferFunction_15779709845963
// MI455X (gfx1250) — compile-verified
//
#include <hip/hip_runtime.h>

typedef float v2f __attribute__((ext_vector_type(2)));
typedef float v8f __attribute__((ext_vector_type(8)));

#define BATCH   16
#define BANDS   128
#define RES     64
#define FRAMES  128
#define SAMPLES 32768
#define ROWS    (BATCH*BANDS)   // 2048 softmax rows

// ---------------------------------------------------------------------------
// Kernel 1: row softmax over RES, then frames = p @ resonance via
// v_wmma_f32_16x16x4_f32.  Block = 16 rows x all 128 frame cols (8 waves,
// one 16x16 N-tile per wave).  Output stored TRANSPOSED per batch:
// framesT[b][f][c]  -> stage 2 can fetch a whole frame column coalesced.
// ---------------------------------------------------------------------------
__global__ __launch_bounds__(256)
void k_softmax_gemm(const float* __restrict__ x,
                    const float* __restrict__ resonance,
                    float* __restrict__ framesT) {
  __shared__ float p[16 * RES];          // 4 KB
  const int tid = threadIdx.x;
  const int m0  = blockIdx.x * 16;       // first global row of this block

  // load 16 rows (1024 floats), coalesced
  #pragma unroll
  for (int k = 0; k < 4; ++k) {
    int idx = tid + k * 256;
    p[idx] = x[m0 * RES + idx];
  }
  __syncthreads();

  // per-row softmax (one thread per row; tiny, latency-irrelevant)
  if (tid < 16) {
    float* row = p + tid * RES;
    float mx = row[0];
    for (int i = 1; i < RES; ++i) mx = fmaxf(mx, row[i]);
    float sum = 0.f;
    for (int i = 0; i < RES; ++i) { float e = __expf(row[i] - mx); row[i] = e; sum += e; }
    float inv = 1.f / sum;
    for (int i = 0; i < RES; ++i) row[i] *= inv;
  }
  __syncthreads();

  const int wave = tid >> 5;
  const int lane = tid & 31;
  const int half = lane >> 4;            // 0: K pair {0,1}; 1: K pair {2,3}
  const int l16  = lane & 15;
  const int n0   = wave * 16;            // frame-column tile

  v8f acc = {};
  for (int kk = 0; kk < RES / 4; ++kk) {
    const int kb = kk * 4 + 2 * half;
    v2f a, b;
    // A 16x4 f32: lane l16 = row M; v0 holds K=kb, v1 holds K=kb+1
    a.x = p[l16 * RES + kb];
    a.y = p[l16 * RES + kb + 1];
    // B 4x16 f32: lane l16 = col N; v0 holds K=kb, v1 holds K=kb+1
    b.x = resonance[kb       * FRAMES + n0 + l16];
    b.y = resonance[(kb + 1) * FRAMES + n0 + l16];
    acc = __builtin_amdgcn_wmma_f32_16x16x4_f32(false, a, false, b,
                                                (short)0, acc, false, false);
  }

  // D 16x16 f32: lane l16 = N col; vgpr v -> M = v + 8*half
  #pragma unroll
  for (int v = 0; v < 8; ++v) {
    int mg = m0 + v + 8 * half;          // global (b,c) row
    int b  = mg >> 7;                    // / BANDS
    int c  = mg & 127;                   // % BANDS
    int f  = n0 + l16;
    framesT[(b * FRAMES + f) * BANDS + c] = acc[v];
  }
}

// ---------------------------------------------------------------------------
// Kernel 2: fused upsample * filter_bank -> mean over bands, as WMMA.
// Each block owns a 128-sample chunk (i0 constant there).  A-tiles pack
// M = (batch, j) rows: A[2b+j][c] = frames[b][c][f0+j].  B = fb[c][s].
// Two accumulators cover batches 0..7 and 8..15; B fragments are shared.
// fb is read exactly once device-wide (16 MB).  Epilogue does the lerp + /128.
// ---------------------------------------------------------------------------
__global__ __launch_bounds__(256)
void k_interp_filter_reduce(const float* __restrict__ framesT,
                            const float* __restrict__ fb,
                            float* __restrict__ out) {
  __shared__ float F0[BATCH * BANDS];    // frames[b][c][f0]  (8 KB)
  __shared__ float F1[BATCH * BANDS];    // frames[b][c][f1]  (8 KB)
  const int tid   = threadIdx.x;
  const int sbase = blockIdx.x * 128;

  // i0 is constant over this 128-sample chunk (boundaries at s = 256f+128)
  float pos0 = fminf(fmaxf(((float)sbase + 0.5f) * (1.0f / 256.0f) - 0.5f,
                           0.0f), 127.0f);
  int f0 = (int)pos0;                    // floor (pos0 >= 0)
  if (f0 > FRAMES - 1) f0 = FRAMES - 1;
  int f1 = f0 + 1; if (f1 > FRAMES - 1) f1 = FRAMES - 1;

  // stage the two needed frame columns, coalesced (framesT rows contiguous)
  #pragma unroll
  for (int k = 0; k < 8; ++k) {
    int idx = tid + k * 256;             // 2048 elems each
    int b = idx >> 7, c = idx & 127;
    F0[idx] = framesT[(b * FRAMES + f0) * BANDS + c];
    F1[idx] = framesT[(b * FRAMES + f1) * BANDS + c];
  }
  __syncthreads();

  const int wave = tid >> 5;
  const int lane = tid & 31;
  const int half = lane >> 4;
  const int l16  = lane & 15;
  const int s0   = sbase + wave * 16;    // this wave's 16-sample tile

  // A row m = l16 maps to (b = m>>1, j = m&1); j selects F0/F1
  const int bi_lo = (l16 >> 1);
  const int bi_hi = 8 + (l16 >> 1);
  const float* Fj = (l16 & 1) ? F1 : F0;

  v8f acc_lo = {}, acc_hi = {};
  for (int kk = 0; kk < BANDS / 4; ++kk) {
    const int kb = kk * 4 + 2 * half;
    v2f bf, a_lo, a_hi;
    bf.x  = fb[kb       * SAMPLES + s0 + l16];   // coalesced 64B / half-wave
    bf.y  = fb[(kb + 1) * SAMPLES + s0 + l16];
    a_lo.x = Fj[bi_lo * BANDS + kb];
    a_lo.y = Fj[bi_lo * BANDS + kb + 1];
    a_hi.x = Fj[bi_hi * BANDS + kb];
    a_hi.y = Fj[bi_hi * BANDS + kb + 1];
    acc_lo = __builtin_amdgcn_wmma_f32_16x16x4_f32(false, a_lo, false, bf,
                                                   (short)0, acc_lo, false, false);
    acc_hi = __builtin_amdgcn_wmma_f32_16x16x4_f32(false, a_hi, false, bf,
                                                   (short)0, acc_hi, false, false);
  }

  // epilogue: per-sample lerp weight, mean over bands
  const int s = s0 + l16;
  float pos = fminf(fmaxf(((float)s + 0.5f) * (1.0f / 256.0f) - 0.5f,
                          0.0f), 127.0f);
  float w = pos - (float)f0;
  const float scale = 1.0f / (float)BANDS;
  #pragma unroll
  for (int v = 0; v < 8; v += 2) {       // even v -> j=0 row, v+1 -> j=1 row
    int m = v + 8 * half;
    float g0 = acc_lo[v], g1 = acc_lo[v + 1];
    out[(m >> 1) * SAMPLES + s]       = (g0 + w * (g1 - g0)) * scale;
    float h0 = acc_hi[v], h1 = acc_hi[v + 1];
    out[(8 + (m >> 1)) * SAMPLES + s] = (h0 + w * (h1 - h0)) * scale;
  }
}

// ---------------------------------------------------------------------------
extern "C" void kernel_launch(void* const* d_in, const int* in_sizes, int n_in,
                              void* d_out, int out_size, void* d_ws, size_t ws_size,
                              hipStream_t stream) {
  (void)in_sizes; (void)n_in; (void)out_size; (void)ws_size;
  const float* x   = (const float*)d_in[0];   // (16,128,64)
  const float* rs  = (const float*)d_in[1];   // (64,128)
  const float* fbk = (const float*)d_in[2];   // (1,128,32768)
  float* out       = (float*)d_out;           // (16,1,32768)
  float* framesT   = (float*)d_ws;            // 16*128*128 f32 = 1 MB scratch

  k_softmax_gemm<<<ROWS / 16, 256, 0, stream>>>(x, rs, framesT);
  k_interp_filter_reduce<<<SAMPLES / 128, 256, 0, stream>>>(framesT, fbk, out);
}